// SimpleDWTClassifier_84945863180807
// MI455X (gfx1250) — compile-verified
//
#include <hip/hip_runtime.h>
#include <hip/hip_bf16.h>

typedef __attribute__((ext_vector_type(2))) float v2f;
typedef __attribute__((ext_vector_type(8))) float v8f;

// ---------------------------------------------------------------------------
// Kernel 1 (tiny, one block): build T = 3-level sym2 wavelet-packet matrix
// (64x50) by pushing the identity through the filter bank, then fold with W
// into Gt (K=152 padded, N=16 padded), stored as float2 k-pairs:
//   g[(p*16 + cls)*2 + par] = dOut[cls]/dX[2p+par]   (zeros outside 150x5)
// ---------------------------------------------------------------------------
__global__ void build_gt_kernel(const float* __restrict__ W,
                                float* __restrict__ g) {
    __shared__ float S1[2 * 26 * 50];   // level-1 bands (len 26)
    __shared__ float S2[4 * 14 * 50];   // level-2 bands (len 14)
    __shared__ float S3[64 * 50];       // level-3 leaves (8 leaves x 8 samples)

    // H0R / H1R = reversed sym2 dec filters (cross-correlation form)
    const float h0[4] = { 0.48296291314453416f,  0.8365163037378079f,
                          0.22414386804185735f, -0.12940952255126037f };
    const float h1[4] = {-0.12940952255126037f, -0.22414386804185735f,
                          0.8365163037378079f,  -0.48296291314453416f };

    const int t = threadIdx.x;
    if (t < 50) {
        // level 1: impulse e_t, len 50 -> 26, ph = 2
        for (int m = 0; m < 26; ++m) {
            float lo = 0.f, hi = 0.f;
            for (int k = 0; k < 4; ++k) {
                int i = 2 * m + k - 2;
                float v = (i == t) ? 1.0f : 0.0f;
                lo += h0[k] * v;  hi += h1[k] * v;
            }
            S1[(0 * 26 + m) * 50 + t] = lo;
            S1[(1 * 26 + m) * 50 + t] = hi;
        }
        // level 2: len 26 -> 14, ph = 2
        for (int b = 0; b < 2; ++b)
            for (int m = 0; m < 14; ++m) {
                float lo = 0.f, hi = 0.f;
                for (int k = 0; k < 4; ++k) {
                    int i = 2 * m + k - 2;
                    float v = (i >= 0 && i < 26) ? S1[(b * 26 + i) * 50 + t] : 0.0f;
                    lo += h0[k] * v;  hi += h1[k] * v;
                }
                S2[((2 * b + 0) * 14 + m) * 50 + t] = lo;
                S2[((2 * b + 1) * 14 + m) * 50 + t] = hi;
            }
        // level 3: len 14 -> 8, ph = 2  (breadth-first lo-first leaf order)
        for (int b = 0; b < 4; ++b)
            for (int m = 0; m < 8; ++m) {
                float lo = 0.f, hi = 0.f;
                for (int k = 0; k < 4; ++k) {
                    int i = 2 * m + k - 2;
                    float v = (i >= 0 && i < 14) ? S2[(b * 14 + i) * 50 + t] : 0.0f;
                    lo += h0[k] * v;  hi += h1[k] * v;
                }
                S3[((2 * b + 0) * 8 + m) * 50 + t] = lo;
                S3[((2 * b + 1) * 8 + m) * 50 + t] = hi;
            }
    }
    __syncthreads();

    // Fold with W. feats layout: feat = leaf*24 + c*8 + s ; W is (5,192) row-major.
    // Fill the entire 76*16*2-float pair buffer (zeros where padded).
    for (int i = t; i < 76 * 16 * 2; i += (int)blockDim.x) {
        const int par = i & 1;
        const int col = (i >> 1) & 15;     // class (N)
        const int p   = i >> 5;            // k-pair
        const int kflat = 2 * p + par;     // flat K = c*50 + n
        float val = 0.0f;
        if (col < 5 && kflat < 150) {
            const int c  = kflat / 50;
            const int kn = kflat % 50;
            for (int f = 0; f < 64; ++f) {
                const int l = f >> 3, s = f & 7;
                val += S3[f * 50 + kn] * W[col * 192 + l * 24 + c * 8 + s];
            }
        }
        g[i] = val;
    }
}

// ---------------------------------------------------------------------------
// Kernel 2: out(Bn x 5) = X(Bn x 150) @ Gt + bias, via V_WMMA_F32_16X16X4_F32.
// One wave32 per 16-row tile; K padded to 152 -> 38 WMMA steps.
// A frag (16x4 f32): lanes 0-15 hold row=lane, K pair {k,k+1};
//                    lanes 16-31 hold row=lane-16, K pair {k+2,k+3}.
// B frag (4x16 f32): symmetric, served from LDS float2 k-pair table.
// ---------------------------------------------------------------------------
__global__ __launch_bounds__(256)
void dwt_wmma_gemm_kernel(const float* __restrict__ X,
                          const float* __restrict__ gB,   // 76*16 float2 pairs
                          const float* __restrict__ bias,
                          float* __restrict__ out, int Bn) {
    __shared__ v2f sB[76 * 16];

    const int tid = threadIdx.x;
    for (int i = tid; i < 76 * 16; i += 256)
        sB[i] = ((const v2f*)gB)[i];
    __syncthreads();

    const int lane    = tid & 31;
    const int wave    = tid >> 5;
    const int tileRow = (blockIdx.x * 8 + wave) * 16;
    const int col     = lane & 15;          // row-in-tile for A / class for B,D
    const int kbase   = (lane >> 4) << 1;   // 0 or 2

    // Clamp row for loads so EXEC stays all-1s through every WMMA.
    int row  = tileRow + col;
    int rowc = row < Bn ? row : (Bn - 1);
    const float* xrow = X + (size_t)rowc * 150;   // 8B-aligned (150*4 % 8 == 0)

    v8f acc = {};
    #pragma unroll
    for (int k = 0; k < 152; k += 4) {
        const int kk = k + kbase;
        const int kc = (kk < 150) ? kk : 0;          // clamped, even -> aligned
        v2f a = *(const v2f*)(xrow + kc);            // global_load_b64
        v2f z = { 0.0f, 0.0f };
        a = (kk < 150) ? a : z;                      // cndmask, no EXEC change
        const v2f bv = sB[(kk >> 1) * 16 + col];     // ds_load_b64
        acc = __builtin_amdgcn_wmma_f32_16x16x4_f32(
            /*neg_a=*/false, a, /*neg_b=*/false, bv,
            /*c_mod=*/(short)0, acc, /*reuse_a=*/false, /*reuse_b=*/false);
    }

    // D layout: VGPR r -> lanes 0-15: (M=r, N=lane); lanes 16-31: (M=r+8, N=lane-16)
    if (col < 5) {
        const float bb = bias[col];
        const int rbase = tileRow + ((lane >> 4) << 3);
        #pragma unroll
        for (int r = 0; r < 8; ++r) {
            const int rr = rbase + r;
            if (rr < Bn) out[(size_t)rr * 5 + col] = acc[r] + bb;
        }
    }
}

// ---------------------------------------------------------------------------
// Launch
// ---------------------------------------------------------------------------
extern "C" void kernel_launch(void* const* d_in, const int* in_sizes, int n_in,
                              void* d_out, int out_size, void* d_ws, size_t ws_size,
                              hipStream_t stream) {
    const float* x    = (const float*)d_in[0];   // (B, 3, 50) f32
    const float* W    = (const float*)d_in[1];   // (5, 192)   f32
    const float* bias = (const float*)d_in[2];   // (5,)       f32
    float*       out  = (float*)d_out;           // (B, 5)     f32
    float*       gws  = (float*)d_ws;            // 76*16*2 floats = 9728 B

    const int Bn = in_sizes[0] / 150;

    build_gt_kernel<<<1, 64, 0, stream>>>(W, gws);

    const int waves  = (Bn + 15) / 16;
    const int blocks = (waves + 7) / 8;
    dwt_wmma_gemm_kernel<<<blocks, 256, 0, stream>>>(x, gws, bias, out, Bn);
}